// Protein_PNAConv_29137058136192
// MI455X (gfx1250) — compile-verified
//
#include <hip/hip_runtime.h>
#include <math.h>

// ---- static problem config (matches reference) ----
#define NN   30000
#define EE   480000
#define HH   128
#define TT   4
#define FF   32
#define EDIM 16
#define AVG_DEG_LOG 2.8332133440562162f   // log(17)

typedef float v2f __attribute__((ext_vector_type(2)));
typedef float v8f __attribute__((ext_vector_type(8)));

// D(16x16) = A(16x4) * B(4x16) + C, fp32 WMMA.
// A layout: lane l: row M = l&15, VGPR0 -> K = kb+2*(l>=16), VGPR1 -> K+1
// B layout: lane l: col N = l&15 (+16*tile), VGPR0 -> row K = kb+2*(l>=16), VGPR1 -> K+1
// C layout: VGPR v: row M = v + 8*(l>=16), col N = (l&15) (+16*tile)
__device__ __forceinline__ v8f wmma4(v2f a, v2f b, v8f c) {
  return __builtin_amdgcn_wmma_f32_16x16x4_f32(
      /*neg_a=*/false, a, /*neg_b=*/false, b,
      /*c_mod=*/(short)0, c, /*reuse_a=*/false, /*reuse_b=*/false);
}

// Wave-local LDS fence: all LDS sharing below is wave-private, so a dscnt
// wait (LDS is in-order within a wave) replaces __syncthreads and avoids the
// s_wait_storecnt 0 drain of in-flight global atomics that a full barrier
// would force.
__device__ __forceinline__ void wave_lds_fence() {
  asm volatile("s_wait_dscnt 0x0" ::: "memory");
}

// Async global->LDS copy (ASYNCcnt-tracked), per-lane 16B.
__device__ __forceinline__ void async_load_b128(unsigned lds_off, const void* gaddr) {
  asm volatile("global_load_async_to_lds_b128 %0, %1, off"
               :: "v"(lds_off), "v"(gaddr) : "memory");
}
__device__ __forceinline__ void wait_async0() {
  asm volatile("s_wait_asynccnt 0x0" ::: "memory");
}

// order-preserving float <-> uint key (unsigned compare order == float order)
__device__ __forceinline__ unsigned f2key(float f) {
  unsigned u = __float_as_uint(f);
  return (u & 0x80000000u) ? ~u : (u | 0x80000000u);
}
__device__ __forceinline__ float key2f(unsigned k) {
  unsigned u = (k & 0x80000000u) ? (k & 0x7FFFFFFFu) : ~k;
  return __uint_as_float(u);
}

// ---------------- kernel 0: init accumulators ----------------
__global__ void pna_init(float* __restrict__ accS, float* __restrict__ accQ,
                         unsigned* __restrict__ accMn, unsigned* __restrict__ accMx,
                         unsigned* __restrict__ cnt) {
  int i = blockIdx.x * blockDim.x + threadIdx.x;
  if (i < NN * HH) {
    accS[i] = 0.f;
    accQ[i] = 0.f;
    accMn[i] = 0xFFFFFFFFu;  // +inf key
    accMx[i] = 0u;           // -inf key
  }
  if (i < NN) cnt[i] = 0u;
}

// ---------------- kernel 1: fused edge encoder + pre-MLP + scatter ----------------
__global__ __launch_bounds__(256) void pna_edge(
    const float* __restrict__ x, const int* __restrict__ ei,
    const float* __restrict__ eattr, const float* __restrict__ We,
    const float* __restrict__ be, const float* __restrict__ W0,
    const float* __restrict__ B0, const float* __restrict__ W1,
    const float* __restrict__ B1,
    float* __restrict__ accS, float* __restrict__ accQ,
    unsigned* __restrict__ accMn, unsigned* __restrict__ accMx,
    unsigned* __restrict__ cnt)
{
  __shared__ float sEA[8][16 * FF];  // encoded edge features per wave tile
  __shared__ float sH[8][16 * FF];   // hidden of pre-MLP
  __shared__ int   sD[8][16];
  __shared__ int   sS[8][16];

  const int wave = threadIdx.x >> 5;
  const int lane = threadIdx.x & 31;
  const int hf   = lane >> 4;   // lane half
  const int l16  = lane & 15;

  const int nTiles     = EE / 16;
  const int wavesPerBk = blockDim.x >> 5;
  const int totalWaves = gridDim.x * wavesPerBk;
  const int iters      = (nTiles + totalWaves - 1) / totalWaves;
  const int tile0      = blockIdx.x * wavesPerBk + wave;

  for (int it = 0; it < iters; ++it) {
    int  tile   = tile0 + it * totalWaves;
    bool active = tile < nTiles;
    int  tc     = active ? tile : 0;
    int  ebase  = tc * 16;

    if (lane < 16) {
      int s = ei[ebase + lane];        // row 0 = src
      int d = ei[EE + ebase + lane];   // row 1 = dst
      sS[wave][lane] = s;
      sD[wave][lane] = d;
      if (active) atomicAdd(&cnt[d], 1u);
      __builtin_prefetch(x + (size_t)d * HH, 0, 3);  // global_prefetch_b8
      __builtin_prefetch(x + (size_t)s * HH, 0, 3);
    }
    wave_lds_fence();

    // --- edge encoder: ea_tile[16x32] = eattr[16x16] @ We[16x32] + be ---
    for (int ct = 0; ct < 2; ++ct) {
      int   col  = ct * 16 + l16;
      float bias = be[col];
      v8f c; for (int i = 0; i < 8; ++i) c[i] = bias;
      size_t arow = (size_t)(ebase + l16) * EDIM;
      for (int kb = 0; kb < EDIM; kb += 4) {
        int k0 = kb + 2 * hf;
        v2f a, b;
        a.x = eattr[arow + k0];
        a.y = eattr[arow + k0 + 1];
        b.x = We[k0 * FF + col];
        b.y = We[(k0 + 1) * FF + col];
        c = wmma4(a, b, c);
      }
      for (int i = 0; i < 8; ++i) sEA[wave][(i + 8 * hf) * FF + col] = c[i];
    }
    wave_lds_fence();

    for (int t = 0; t < TT; ++t) {
      // --- pre-MLP layer 0: h = relu([x_dst | x_src | ea](16x96) @ W0t(96x32) + b0) ---
      const float* W0t = W0 + (size_t)t * (3 * FF) * FF;
      int dn = sD[wave][l16];
      int sn = sS[wave][l16];
      for (int ct = 0; ct < 2; ++ct) {
        int   col  = ct * 16 + l16;
        float bias = B0[t * FF + col];
        v8f c; for (int i = 0; i < 8; ++i) c[i] = bias;
        for (int kb = 0; kb < 3 * FF; kb += 4) {
          int k0 = kb + 2 * hf;
          v2f a, b;
          if (k0 < FF) {
            a.x = x[(size_t)dn * HH + t * FF + k0];
            a.y = x[(size_t)dn * HH + t * FF + k0 + 1];
          } else if (k0 < 2 * FF) {
            a.x = x[(size_t)sn * HH + t * FF + (k0 - FF)];
            a.y = x[(size_t)sn * HH + t * FF + (k0 - FF) + 1];
          } else {
            a.x = sEA[wave][l16 * FF + (k0 - 2 * FF)];
            a.y = sEA[wave][l16 * FF + (k0 - 2 * FF) + 1];
          }
          b.x = W0t[k0 * FF + col];
          b.y = W0t[(k0 + 1) * FF + col];
          c = wmma4(a, b, c);
        }
        for (int i = 0; i < 8; ++i)
          sH[wave][(i + 8 * hf) * FF + col] = fmaxf(c[i], 0.f);
      }
      wave_lds_fence();

      // --- pre-MLP layer 1 + 4-way atomic scatter ---
      const float* W1t = W1 + (size_t)t * FF * FF;
      for (int ct = 0; ct < 2; ++ct) {
        int   col  = ct * 16 + l16;
        float bias = B1[t * FF + col];
        v8f c; for (int i = 0; i < 8; ++i) c[i] = bias;
        for (int kb = 0; kb < FF; kb += 4) {
          int k0 = kb + 2 * hf;
          v2f a, b;
          a.x = sH[wave][l16 * FF + k0];
          a.y = sH[wave][l16 * FF + k0 + 1];
          b.x = W1t[k0 * FF + col];
          b.y = W1t[(k0 + 1) * FF + col];
          c = wmma4(a, b, c);
        }
        if (active) {
          int ch = t * FF + col;
          for (int i = 0; i < 8; ++i) {
            int    r    = i + 8 * hf;        // edge row in tile
            int    node = sD[wave][r];
            size_t idx  = (size_t)node * HH + ch;
            float  v    = c[i];
            __hip_atomic_fetch_add(&accS[idx], v, __ATOMIC_RELAXED,
                                   __HIP_MEMORY_SCOPE_AGENT);
            __hip_atomic_fetch_add(&accQ[idx], v * v, __ATOMIC_RELAXED,
                                   __HIP_MEMORY_SCOPE_AGENT);
            unsigned key = f2key(v);
            atomicMin(&accMn[idx], key);
            atomicMax(&accMx[idx], key);
          }
        }
      }
      wave_lds_fence();
    }
  }
}

// ---------------- kernel 2: finalize stats (in place) + degree scalers ----------------
__global__ void pna_finalize(const unsigned* __restrict__ cnt,
                             const float* __restrict__ accS,
                             float* __restrict__ accQ,
                             unsigned* __restrict__ accMn,
                             unsigned* __restrict__ accMx,
                             float* __restrict__ stdb,
                             float* __restrict__ scal)
{
  int i = blockIdx.x * blockDim.x + threadIdx.x;
  if (i >= NN * HH) return;
  int   n     = i / HH;
  float cf    = (float)cnt[n];
  float denom = fmaxf(cf, 1.f);
  float mean  = accS[i] / denom;
  float m2    = accQ[i] / denom;
  float var   = fmaxf(m2 - mean * mean, 0.f);
  float sd    = sqrtf(var + 1e-5f);
  bool  has   = cf > 0.f;
  float mn    = has ? key2f(accMn[i]) : 0.f;
  float mx    = has ? key2f(accMx[i]) : 0.f;
  accQ[i] = mean;                    // kind 1
  ((float*)accMn)[i] = mn;           // kind 2
  ((float*)accMx)[i] = mx;           // kind 3
  stdb[i] = sd;                      // kind 4  (kind 0 = accS, untouched)
  if ((i % HH) == 0) {
    float logd = logf(denom + 1.f);
    scal[2 * n]     = logd * (1.f / AVG_DEG_LOG);  // amplification
    scal[2 * n + 1] = AVG_DEG_LOG / logd;          // attenuation
  }
}

// post-MLP input column c (0..511) for node n, tower t
__device__ __forceinline__ float post_col(
    int n, int t, int c,
    const float* __restrict__ x, const float* __restrict__ fS,
    const float* __restrict__ fMean, const float* __restrict__ fMn,
    const float* __restrict__ fMx, const float* __restrict__ fSd,
    float amp, float att)
{
  if (c < FF) return x[(size_t)n * HH + t * FF + c];
  int cc     = c - FF;
  int grp    = cc / (5 * FF);
  int within = cc - grp * (5 * FF);
  int kind   = within / FF;
  int ch     = t * FF + (within - kind * FF);
  size_t idx = (size_t)n * HH + ch;
  float v;
  switch (kind) {
    case 0:  v = fS[idx];    break;
    case 1:  v = fMean[idx]; break;
    case 2:  v = fMn[idx];   break;
    case 3:  v = fMx[idx];   break;
    default: v = fSd[idx];   break;
  }
  float sc = (grp == 0) ? 1.f : ((grp == 1) ? amp : att);
  return v * sc;
}

// ---------------- kernel 3: per-tower post-MLP ----------------
__global__ __launch_bounds__(256) void pna_post(
    const float* __restrict__ x,
    const float* __restrict__ fS, const float* __restrict__ fMean,
    const float* __restrict__ fMn, const float* __restrict__ fMx,
    const float* __restrict__ fSd, const float* __restrict__ scal,
    const float* __restrict__ W0, const float* __restrict__ B0,
    const float* __restrict__ W1, const float* __restrict__ B1,
    float* __restrict__ hbuf)
{
  __shared__ float sH[8][16 * FF];
  __shared__ float sAmp[8][16];
  __shared__ float sAtt[8][16];

  const int wave = threadIdx.x >> 5;
  const int lane = threadIdx.x & 31;
  const int hf   = lane >> 4;
  const int l16  = lane & 15;
  const int KIN  = (5 * 3 + 1) * FF;  // 512

  const int nTiles     = NN / 16;
  const int wavesPerBk = blockDim.x >> 5;
  const int totalWaves = gridDim.x * wavesPerBk;
  const int iters      = (nTiles + totalWaves - 1) / totalWaves;
  const int tile0      = blockIdx.x * wavesPerBk + wave;

  for (int it = 0; it < iters; ++it) {
    int  tile   = tile0 + it * totalWaves;
    bool active = tile < nTiles;
    int  tc     = active ? tile : 0;
    int  nbase  = tc * 16;

    if (lane < 16) {
      sAmp[wave][lane] = scal[2 * (nbase + lane)];
      sAtt[wave][lane] = scal[2 * (nbase + lane) + 1];
    }
    wave_lds_fence();

    int   myrow = nbase + l16;
    float amp   = sAmp[wave][l16];
    float att   = sAtt[wave][l16];

    for (int t = 0; t < TT; ++t) {
      const float* W0t = W0 + (size_t)t * KIN * FF;
      for (int ct = 0; ct < 2; ++ct) {
        int   col  = ct * 16 + l16;
        float bias = B0[t * FF + col];
        v8f c; for (int i = 0; i < 8; ++i) c[i] = bias;
        for (int kb = 0; kb < KIN; kb += 4) {
          int k0 = kb + 2 * hf;
          v2f a, b;
          a.x = post_col(myrow, t, k0,     x, fS, fMean, fMn, fMx, fSd, amp, att);
          a.y = post_col(myrow, t, k0 + 1, x, fS, fMean, fMn, fMx, fSd, amp, att);
          b.x = W0t[(size_t)k0 * FF + col];
          b.y = W0t[(size_t)(k0 + 1) * FF + col];
          c = wmma4(a, b, c);
        }
        for (int i = 0; i < 8; ++i)
          sH[wave][(i + 8 * hf) * FF + col] = fmaxf(c[i], 0.f);
      }
      wave_lds_fence();

      const float* W1t = W1 + (size_t)t * FF * FF;
      for (int ct = 0; ct < 2; ++ct) {
        int   col  = ct * 16 + l16;
        float bias = B1[t * FF + col];
        v8f c; for (int i = 0; i < 8; ++i) c[i] = bias;
        for (int kb = 0; kb < FF; kb += 4) {
          int k0 = kb + 2 * hf;
          v2f a, b;
          a.x = sH[wave][l16 * FF + k0];
          a.y = sH[wave][l16 * FF + k0 + 1];
          b.x = W1t[k0 * FF + col];
          b.y = W1t[(k0 + 1) * FF + col];
          c = wmma4(a, b, c);
        }
        if (active) {
          for (int i = 0; i < 8; ++i) {
            int node = nbase + i + 8 * hf;
            hbuf[(size_t)node * HH + t * FF + col] = c[i];
          }
        }
      }
      wave_lds_fence();
    }
  }
}

// ---------------- kernel 4: final linear + LayerNorm + residual ReLU ----------------
// A tile (16 consecutive hbuf rows = contiguous 8KB) is staged once into LDS
// via async global->LDS B128 copies, then consumed 8x (one per column tile).
// LayerNorm is done entirely in registers with shfl_xor reductions over the
// 16-lane halves (matches C-layout row grouping on wave32).
__global__ __launch_bounds__(128) void pna_final(
    const float* __restrict__ x, const float* __restrict__ hbuf,
    const float* __restrict__ Wl, const float* __restrict__ bl,
    const float* __restrict__ g, const float* __restrict__ bb,
    float* __restrict__ out)
{
  __shared__ float sA[4][16 * HH] __attribute__((aligned(16)));

  const int wave = threadIdx.x >> 5;
  const int lane = threadIdx.x & 31;
  const int hf   = lane >> 4;
  const int l16  = lane & 15;

  const int nTiles     = NN / 16;
  const int wavesPerBk = blockDim.x >> 5;
  const int totalWaves = gridDim.x * wavesPerBk;
  const int iters      = (nTiles + totalWaves - 1) / totalWaves;
  const int tile0      = blockIdx.x * wavesPerBk + wave;

  const unsigned lbase = (unsigned)(size_t)(&sA[wave][0]);

  for (int it = 0; it < iters; ++it) {
    int  tile   = tile0 + it * totalWaves;
    bool active = tile < nTiles;
    int  tc     = active ? tile : 0;
    int  nbase  = tc * 16;

    // make sure last iteration's LDS reads retired before async overwrite
    wave_lds_fence();

    // async stage of the 16x128 A tile (8KB contiguous): 16 x (32 lanes x 16B)
    const char* gp   = (const char*)(hbuf + (size_t)nbase * HH) + lane * 16;
    unsigned    loff = lbase + (unsigned)lane * 16u;
    for (int s = 0; s < 16; ++s)
      async_load_b128(loff + (unsigned)s * 512u, gp + (size_t)s * 512);
    wait_async0();

    v8f acc[8];
    for (int ct = 0; ct < 8; ++ct) {
      int   col  = ct * 16 + l16;
      float bias = bl[col];
      v8f c; for (int i = 0; i < 8; ++i) c[i] = bias;
      for (int kb = 0; kb < HH; kb += 4) {
        int k0 = kb + 2 * hf;
        v2f a, b;
        a.x = sA[wave][l16 * HH + k0];
        a.y = sA[wave][l16 * HH + k0 + 1];
        b.x = Wl[(size_t)k0 * HH + col];
        b.y = Wl[(size_t)(k0 + 1) * HH + col];
        c = wmma4(a, b, c);
      }
      acc[ct] = c;
    }

    // per-row LayerNorm + residual ReLU, fully in registers
    for (int i = 0; i < 8; ++i) {
      float s = 0.f, sq = 0.f;
      for (int ct = 0; ct < 8; ++ct) {
        float v = acc[ct][i];
        s += v; sq += v * v;
      }
      // reduce over the 16 lanes of this half (row is fixed within half)
      for (int m = 1; m <= 8; m <<= 1) {
        s  += __shfl_xor(s, m, 32);
        sq += __shfl_xor(sq, m, 32);
      }
      float mu  = s * (1.f / HH);
      float var = fmaxf(sq * (1.f / HH) - mu * mu, 0.f);
      float rs  = rsqrtf(var + 1e-5f);
      if (active) {
        int n = nbase + i + 8 * hf;
        for (int ct = 0; ct < 8; ++ct) {
          int   cidx = ct * 16 + l16;
          float v    = acc[ct][i];
          float ln   = (v - mu) * rs * g[cidx] + bb[cidx];
          out[(size_t)n * HH + cidx] = x[(size_t)n * HH + cidx] + fmaxf(ln, 0.f);
        }
      }
    }
  }
}

extern "C" void kernel_launch(void* const* d_in, const int* in_sizes, int n_in,
                              void* d_out, int out_size, void* d_ws, size_t ws_size,
                              hipStream_t stream)
{
  (void)in_sizes; (void)n_in; (void)out_size; (void)ws_size;

  const float* x      = (const float*)d_in[0];
  const int*   ei     = (const int*)d_in[1];
  const float* eattr  = (const float*)d_in[2];
  const float* We     = (const float*)d_in[3];
  const float* be     = (const float*)d_in[4];
  const float* preW0  = (const float*)d_in[5];
  const float* preB0  = (const float*)d_in[6];
  const float* preW1  = (const float*)d_in[7];
  const float* preB1  = (const float*)d_in[8];
  const float* postW0 = (const float*)d_in[9];
  const float* postB0 = (const float*)d_in[10];
  const float* postW1 = (const float*)d_in[11];
  const float* postB1 = (const float*)d_in[12];
  const float* Wl     = (const float*)d_in[13];
  const float* bl     = (const float*)d_in[14];
  const float* lng    = (const float*)d_in[15];
  const float* lnb    = (const float*)d_in[16];
  float* out = (float*)d_out;

  char* w = (char*)d_ws;
  auto alloc = [&](size_t bytes) {
    char* p = w;
    w += (bytes + 255) & ~(size_t)255;
    return p;
  };
  float*    accS  = (float*)alloc((size_t)NN * HH * 4);
  float*    accQ  = (float*)alloc((size_t)NN * HH * 4);
  unsigned* accMn = (unsigned*)alloc((size_t)NN * HH * 4);
  unsigned* accMx = (unsigned*)alloc((size_t)NN * HH * 4);
  float*    stdb  = (float*)alloc((size_t)NN * HH * 4);
  unsigned* cnt   = (unsigned*)alloc((size_t)NN * 4);
  float*    scal  = (float*)alloc((size_t)NN * 2 * 4);
  float*    hbuf  = (float*)alloc((size_t)NN * HH * 4);

  // 0) init accumulators (must run every call; ws is not re-zeroed)
  pna_init<<<(NN * HH + 255) / 256, 256, 0, stream>>>(accS, accQ, accMn, accMx, cnt);

  // 1) fused edge encoder + pre-MLP + scatter reduce (1 wave per 16-edge tile)
  pna_edge<<<EE / 16 / 8, 256, 0, stream>>>(x, ei, eattr, We, be,
                                            preW0, preB0, preW1, preB1,
                                            accS, accQ, accMn, accMx, cnt);

  // 2) finalize mean/min/max/std + degree scalers
  pna_finalize<<<(NN * HH + 255) / 256, 256, 0, stream>>>(cnt, accS, accQ, accMn,
                                                          accMx, stdb, scal);

  // 3) per-tower post-MLP (1 wave per 16-node tile)
  pna_post<<<(NN / 16 + 7) / 8, 256, 0, stream>>>(
      x, accS, accQ, (const float*)accMn, (const float*)accMx, stdb, scal,
      postW0, postB0, postW1, postB1, hbuf);

  // 4) final linear + LayerNorm + residual ReLU (async A-tile staging)
  pna_final<<<(NN / 16 + 3) / 4, 128, 0, stream>>>(x, hbuf, Wl, bl, lng, lnb, out);
}